// AdvDiffSolver_687194767683
// MI455X (gfx1250) — compile-verified
//
#include <hip/hip_runtime.h>
#include <stdint.h>

// Advection-diffusion: y' = d*y_xx/DX^2 - c*y_x/(2DX), DT=DX=0.01
// => y_new[i] = y[i] + (100*d)*(y[i-1]-2y[i]+y[i+1]) - (0.5*c)*(y[i+1]-y[i-1])
// Output x[b][i][t], t=0..999 (t=0 = initial condition).

constexpr int NX    = 1024;
constexpr int NT    = 1000;
constexpr int WAVES = 16;
constexpr int LANES = 32;          // wave32 on gfx1250
constexpr int OWN   = NX / WAVES;  // 64 points owned per wave
constexpr int HALO  = 16;          // ghost cells per side, resync every 16 steps
constexpr int EXT   = OWN + 2 * HALO;  // 96 extended cells per wave
constexpr int PER   = EXT / LANES;     // 3 cells per lane (register-resident)

__global__ __launch_bounds__(WAVES * LANES)
void adv_diff_halo_kernel(const float* __restrict__ init,
                          const float* __restrict__ params,
                          float* __restrict__ out)
{
    __shared__ float pub[NX];  // init staging + halo publish buffer (4 KB)

    const int tid = threadIdx.x;
    const int w   = tid >> 5;   // wave within block
    const int l   = tid & 31;   // lane within wave
    const int b   = blockIdx.x; // batch row

    // Per-row coefficients (uniform per block -> scalar loads)
    const float c1 = params[2 * b + 0] * 100.0f; // DT/DX^2
    const float c2 = params[2 * b + 1] * 0.5f;   // DT/(2*DX)

    // ---- CDNA5 async global->LDS copy of the initial row (ASYNCcnt path) ----
    {
        const float* gsrc   = init + (size_t)b * NX + (size_t)tid * 2;
        unsigned     ldsoff = (unsigned)(uintptr_t)(&pub[tid * 2]);
        asm volatile("global_load_async_to_lds_b64 %0, %1, off"
                     :: "v"(ldsoff), "v"(gsrc) : "memory");
        asm volatile("s_wait_asynccnt 0x0" ::: "memory");
    }
    __syncthreads();

    // ---- static per-cell metadata ----
    int    p[PER];                 // global spatial index of each cell
    float  mask[PER];              // 1.0 in-domain, 0.0 for zero-pad ghosts
    float  c1v[PER], c2v[PER];     // coeffs with boundary mask folded in:
                                   // out-of-domain cells get c=0 => y stays 0
    bool   owned[PER], halo[PER], pubc[PER];
    float* optr[PER];              // running output pointer (advances 4 floats/flush)
    const int cbase = l * PER;
    #pragma unroll
    for (int r = 0; r < PER; ++r) {
        const int c = cbase + r;               // extended-cell index 0..EXT-1
        p[r]     = w * OWN - HALO + c;         // global position (may be out of domain)
        mask[r]  = (p[r] >= 0 && p[r] < NX) ? 1.0f : 0.0f;
        c1v[r]   = c1 * mask[r];
        c2v[r]   = c2 * mask[r];
        owned[r] = (c >= HALO) && (c < HALO + OWN);
        halo[r]  = !owned[r];
        pubc[r]  = owned[r] && ((c < 2 * HALO) || (c >= OWN)); // 16 edge cells/side
        optr[r]  = out + (long long)(b * NX + p[r]) * NT;
    }

    // ---- load state from LDS ----
    float y[PER];
    #pragma unroll
    for (int r = 0; r < PER; ++r) {
        int q = p[r] < 0 ? 0 : (p[r] >= NX ? NX - 1 : p[r]);
        y[r] = pub[q] * mask[r];
    }
    __syncthreads();  // everyone done with init staging before pub is reused

    float4 ring[PER];  // 4-step time ring per cell -> b128 streaming stores

    auto step = [&]() {
        // neighbors: in-lane register shifts + 2 wave shuffles for lane edges
        const float up = __shfl_up(y[PER - 1], 1, 32);  // lane l-1's last cell
        const float dn = __shfl_down(y[0], 1, 32);      // lane l+1's first cell
        float yl[PER], yr[PER];
        yl[0] = up;
        #pragma unroll
        for (int r = 1; r < PER; ++r) yl[r] = y[r - 1];
        #pragma unroll
        for (int r = 0; r < PER - 1; ++r) yr[r] = y[r + 1];
        yr[PER - 1] = dn;
        #pragma unroll
        for (int r = 0; r < PER; ++r) {
            const float s   = yl[r] + yr[r];
            const float lap = __builtin_fmaf(-2.0f, y[r], s);   // y_xx stencil
            const float adv = yr[r] - yl[r];                    // y_x stencil
            float yn = __builtin_fmaf(c1v[r], lap, y[r]);
            y[r]     = __builtin_fmaf(-c2v[r], adv, yn);
        }
    };

    auto record = [&](int s) {  // s is always a literal -> constant-folded
        #pragma unroll
        for (int r = 0; r < PER; ++r) (&ring[r].x)[s] = y[r];
    };

    auto flush = [&]() {  // one 16B store per owned cell; L2 merges to full lines
        #pragma unroll
        for (int r = 0; r < PER; ++r) {
            if (owned[r]) *(float4*)optr[r] = ring[r];
            optr[r] += 4;
        }
    };

    auto refresh = [&]() {  // re-validate ghost cells from neighbor waves
        __syncthreads();
        #pragma unroll
        for (int r = 0; r < PER; ++r)
            if (pubc[r]) pub[p[r]] = y[r];
        __syncthreads();
        #pragma unroll
        for (int r = 0; r < PER; ++r)
            if (halo[r]) {
                int q = p[r] < 0 ? 0 : (p[r] >= NX ? NX - 1 : p[r]);
                y[r] = pub[q] * mask[r];
            }
    };

    // group 0: t=0 is the initial condition, then steps t=1..3
    record(0);
    step(); record(1);
    step(); record(2);
    step(); record(3);
    flush();

    // groups 1..249: steps t=4g..4g+3; ghost resync every 16 steps (after t=15,31,...)
    for (int g = 1; g < 250; ++g) {
        step(); record(0);
        step(); record(1);
        step(); record(2);
        step(); record(3);
        flush();
        if ((g & 3) == 3 && g != 249) refresh();
    }
}

extern "C" void kernel_launch(void* const* d_in, const int* in_sizes, int n_in,
                              void* d_out, int out_size, void* d_ws, size_t ws_size,
                              hipStream_t stream) {
    const float* init   = (const float*)d_in[0];  // (64, 1024) f32
    const float* params = (const float*)d_in[1];  // (64, 2)    f32
    float*       out    = (float*)d_out;          // (64, 1024, 1000) f32

    const int batches = in_sizes[0] / NX;  // 64
    adv_diff_halo_kernel<<<dim3(batches), dim3(WAVES * LANES), 0, stream>>>(
        init, params, out);
}